// MyEdgeConv_32014686224670
// MI455X (gfx1250) — compile-verified
//
#include <hip/hip_runtime.h>

typedef float v2f __attribute__((ext_vector_type(2)));
typedef float v8f __attribute__((ext_vector_type(8)));

#define NN 50000
#define NE 800000
#define DD 64

// Native CDNA5 no-return float max atomic at device scope (L2 RMW unit).
__device__ __forceinline__ void atom_max_f32(float* p, float v) {
    asm volatile("global_atomic_max_num_f32 %0, %1, off scope:SCOPE_DEV"
                 :: "v"(p), "v"(v) : "memory");
}

// ---------------- Kernel 1: fill output with -inf ----------------
__global__ void ec_init(float* __restrict__ out) {
    int t = blockIdx.x * blockDim.x + threadIdx.x;   // NN*DD/4 threads
    const float ni = -__builtin_huge_valf();
    ((float4*)out)[t] = make_float4(ni, ni, ni, ni);
}

// ---------------- Kernel 2: node-level GEMMs via fp32 WMMA ----------------
// A'[n] = feat[n] @ theta_w.T + (theta_b + phi_b)
// B [n] = feat[n] @ (phi_w - theta_w).T
// Block = 128 threads = 4 waves; wave w computes output cols [16w,16w+16) for
// a 16-row node tile. K looped 0..63 step 4 with V_WMMA_F32_16X16X4_F32.
__global__ void ec_gemm(const float* __restrict__ feat,
                        const float* __restrict__ theta_w,
                        const float* __restrict__ theta_b,
                        const float* __restrict__ phi_w,
                        const float* __restrict__ phi_b,
                        float* __restrict__ Aout,
                        float* __restrict__ Bout) {
    const int lane = threadIdx.x & 31;
    const int wave = threadIdx.x >> 5;      // 0..3 -> N tile
    const int l15  = lane & 15;
    const int h    = lane >> 4;             // lane half
    const int base = blockIdx.x * 16;       // node row tile
    const int n0   = wave * 16;             // output column tile

    const float* frow = feat + (size_t)(base + l15) * DD;
    const int    wr   = n0 + l15;           // weight row = output column

    v8f acc_t = {};   // theta accumulator
    v8f acc_c = {};   // (phi - theta) accumulator

#pragma unroll
    for (int k0 = 0; k0 < DD; k0 += 4) {
        const int kc = k0 + 2 * h;          // fragment K per ISA layout: K = r + 2h
        // A fragment: feat[base+l15][kc], [kc+1]  (contiguous -> 8B load)
        v2f a = *(const v2f*)(frow + kc);
        // B fragments: W.T[kc + r][wr] == W[wr][kc + r]  (contiguous -> 8B load)
        v2f bt = *(const v2f*)(theta_w + wr * DD + kc);
        v2f bp = *(const v2f*)(phi_w   + wr * DD + kc);
        v2f bc = bp - bt;
        acc_t = __builtin_amdgcn_wmma_f32_16x16x4_f32(
            false, a, false, bt, (short)0, acc_t, false, false);
        acc_c = __builtin_amdgcn_wmma_f32_16x16x4_f32(
            false, a, false, bc, (short)0, acc_c, false, false);
    }

    const float bias = theta_b[wr] + phi_b[wr];
#pragma unroll
    for (int r = 0; r < 8; ++r) {
        const int m = base + r + 8 * h;     // C/D layout: M = r + 8h, N = lane 0..15
        Aout[(size_t)m * DD + n0 + l15] = acc_t[r] + bias;
        Bout[(size_t)m * DD + n0 + l15] = acc_c[r];
    }
}

// ---------------- Kernel 3: edge scatter-max ----------------
// out[dst] = max(out[dst], B[src])  elementwise; 16 threads per edge x float4.
__global__ void ec_scatter(const int* __restrict__ src,
                           const int* __restrict__ dst,
                           const float* __restrict__ Bw,
                           float* __restrict__ out) {
    int t = blockIdx.x * blockDim.x + threadIdx.x;   // NE*16 threads
    int e = t >> 4;
    int c = (t & 15) * 4;
    int s = src[e];
    int d = dst[e];
    float4 b = *(const float4*)(Bw + (size_t)s * DD + c);
    float* o = out + (size_t)d * DD + c;
    atom_max_f32(o + 0, b.x);
    atom_max_f32(o + 1, b.y);
    atom_max_f32(o + 2, b.z);
    atom_max_f32(o + 3, b.w);
}

// ---------------- Kernel 4: finalize ----------------
// out = (out == -inf) ? 0 : out + A'
__global__ void ec_final(float* __restrict__ out, const float* __restrict__ Ap) {
    int t = blockIdx.x * blockDim.x + threadIdx.x;   // NN*DD/4 threads
    const float ni = -__builtin_huge_valf();
    float4 m = ((const float4*)out)[t];
    float4 a = ((const float4*)Ap)[t];
    float4 r;
    r.x = (m.x == ni) ? 0.0f : m.x + a.x;
    r.y = (m.y == ni) ? 0.0f : m.y + a.y;
    r.z = (m.z == ni) ? 0.0f : m.z + a.z;
    r.w = (m.w == ni) ? 0.0f : m.w + a.w;
    ((float4*)out)[t] = r;
}

extern "C" void kernel_launch(void* const* d_in, const int* in_sizes, int n_in,
                              void* d_out, int out_size, void* d_ws, size_t ws_size,
                              hipStream_t stream) {
    const float* feat    = (const float*)d_in[0];
    const float* theta_w = (const float*)d_in[1];
    const float* theta_b = (const float*)d_in[2];
    const float* phi_w   = (const float*)d_in[3];
    const float* phi_b   = (const float*)d_in[4];
    const int*   src     = (const int*)d_in[5];
    const int*   dst     = (const int*)d_in[6];
    float* out = (float*)d_out;

    float* Aout = (float*)d_ws;                    // 50000*64 f32 = 12.8 MB
    float* Bout = Aout + (size_t)NN * DD;          // 50000*64 f32 = 12.8 MB

    // 1) out <- -inf        (3.2M elems / 4 per thread)
    ec_init<<<(NN * DD / 4) / 256, 256, 0, stream>>>(out);
    // 2) node GEMMs (WMMA)  (3125 row tiles, 4 waves each)
    ec_gemm<<<NN / 16, 128, 0, stream>>>(feat, theta_w, theta_b, phi_w, phi_b,
                                         Aout, Bout);
    // 3) edge scatter-max   (800K edges x 16 threads)
    ec_scatter<<<(NE * 16) / 256, 256, 0, stream>>>(src, dst, Bout, out);
    // 4) finalize
    ec_final<<<(NN * DD / 4) / 256, 256, 0, stream>>>(out, Aout);
}